// nativeContrastLoss_subclass_24876450578883
// MI455X (gfx1250) — compile-verified
//
#include <hip/hip_runtime.h>
#include <math.h>

// ---- problem constants (from reference) ----
#define NA      4096          // N_ANCHOR
#define DIMV    64            // DIM
#define NQ      15300         // NUM_CLASSES*K*PIXEL_SIZE
#define NQP     15312         // NQ padded to multiple of 16
#define NCC     102           // NUM_CLASSES*K
#define NCCP    112           // padded to multiple of 16
#define PIX     150
#define INV_T   10.0f         // 1/TEMP

typedef float v2f __attribute__((ext_vector_type(2)));
typedef float v8f __attribute__((ext_vector_type(8)));

// -------- row L2 normalization (one thread per row) --------
__global__ __launch_bounds__(256) void norm_rows(const float* __restrict__ src,
                                                 float* __restrict__ dst,
                                                 int nrows, int totrows) {
  int r = blockIdx.x * blockDim.x + threadIdx.x;
  if (r >= totrows) return;
  float* d = dst + (size_t)r * DIMV;
  if (r < nrows) {
    const float* s = src + (size_t)r * DIMV;
    float ss = 0.f;
    #pragma unroll 8
    for (int k = 0; k < DIMV; ++k) { float v = s[k]; ss += v * v; }
    float inv = 1.0f / fmaxf(sqrtf(ss), 1e-12f);
    #pragma unroll 8
    for (int k = 0; k < DIMV; ++k) d[k] = s[k] * inv;
  } else {
    #pragma unroll 8
    for (int k = 0; k < DIMV; ++k) d[k] = 0.f;   // zero pad rows
  }
}

// -------- WMMA helpers --------
// A (16x4 f32): lanes 0-15 -> M=lane, K = 4c + {0,1}; lanes 16-31 -> M=lane-16, K = 4c + {2,3}
// B (4x16 f32): lanes 0-15 -> N=lane, K = 4c + {0,1}; lanes 16-31 -> N=lane-16, K = 4c + {2,3}
// C (16x16 f32): lane L, VGPR r -> M = r + 8*(L>>4), N = L&15
__device__ __forceinline__ v8f gemm16(const v2f* a, const float* __restrict__ Bbase,
                                      int colBase, int lm, int half) {
  const float* brow = Bbase + (size_t)(colBase + lm) * DIMV + 2 * half;
  v8f acc = {};
  #pragma unroll
  for (int c = 0; c < 16; ++c) {
    v2f b = *(const v2f*)(brow + 4 * c);
    acc = __builtin_amdgcn_wmma_f32_16x16x4_f32(false, a[c], false, b,
                                                (short)0, acc, false, false);
  }
  return acc;
}

__device__ __forceinline__ float redmax16(float x) {
  #pragma unroll
  for (int m = 1; m < 16; m <<= 1) x = fmaxf(x, __shfl_xor(x, m, 32));
  return x;
}
__device__ __forceinline__ float redsum16(float x) {
  #pragma unroll
  for (int m = 1; m < 16; m <<= 1) x += __shfl_xor(x, m, 32);
  return x;
}

// -------- main per-anchor statistics: one wave handles 16 anchor rows --------
__global__ __launch_bounds__(32) void stats_kernel(
    const float* __restrict__ Xn, const float* __restrict__ Qn,
    const float* __restrict__ Cn, const int* __restrict__ y,
    float* __restrict__ per_ppc, float* __restrict__ validArr,
    float* __restrict__ mlppc) {
  const int lane = threadIdx.x & 31;
  const int half = lane >> 4;
  const int lm   = lane & 15;
  const int rowBase = blockIdx.x * 16;

  // resident A fragments: this wave's 16x64 fp32 anchor tile
  v2f a[16];
  {
    const float* arow = Xn + (size_t)(rowBase + lm) * DIMV + 2 * half;
    #pragma unroll
    for (int c = 0; c < 16; ++c) a[c] = *(const v2f*)(arow + 4 * c);
  }
  int yrow[8];
  #pragma unroll
  for (int r = 0; r < 8; ++r) yrow[r] = y[rowBase + r + 8 * half];

  // ---- pass 1a: row max over adc (all 4096 cols, diag included) ----
  float m1[8];
  {
    float pm[8];
    #pragma unroll
    for (int r = 0; r < 8; ++r) pm[r] = -3.4e38f;
    for (int t = 0; t < NA / 16; ++t) {
      v8f acc = gemm16(a, Xn, t * 16, lm, half);
      #pragma unroll
      for (int r = 0; r < 8; ++r) pm[r] = fmaxf(pm[r], acc[r] * INV_T);
    }
    #pragma unroll
    for (int r = 0; r < 8; ++r) m1[r] = redmax16(pm[r]);
  }

  // ---- pass 1b: row max over adc_q ----
  float m2[8];
  {
    float pm[8];
    #pragma unroll
    for (int r = 0; r < 8; ++r) pm[r] = -3.4e38f;
    for (int t = 0; t < NQP / 16; ++t) {
      int j = t * 16 + lm;
      v8f acc = gemm16(a, Qn, t * 16, lm, half);
      if (j < NQ) {
        #pragma unroll
        for (int r = 0; r < 8; ++r) pm[r] = fmaxf(pm[r], acc[r] * INV_T);
      }
    }
    #pragma unroll
    for (int r = 0; r < 8; ++r) m2[r] = redmax16(pm[r]);
  }

  // ---- pass 2a: neg_logits_raw ----
  float Dr[8];
  {
    float nr[8] = {0.f, 0.f, 0.f, 0.f, 0.f, 0.f, 0.f, 0.f};
    for (int t = 0; t < NA / 16; ++t) {
      int ycol = y[t * 16 + lm];
      v8f acc = gemm16(a, Xn, t * 16, lm, half);
      #pragma unroll
      for (int r = 0; r < 8; ++r)
        if (ycol != yrow[r]) nr[r] += __expf(acc[r] * INV_T - m1[r]);
    }
    #pragma unroll
    for (int r = 0; r < 8; ++r) Dr[r] = redsum16(nr[r]);
  }

  // ---- pass 2b: neg_logits_q ----
  {
    float nq[8] = {0.f, 0.f, 0.f, 0.f, 0.f, 0.f, 0.f, 0.f};
    for (int t = 0; t < NQP / 16; ++t) {
      int j = t * 16 + lm;
      int yq = j / PIX;
      bool jv = j < NQ;
      v8f acc = gemm16(a, Qn, t * 16, lm, half);
      #pragma unroll
      for (int r = 0; r < 8; ++r)
        if (jv && yq != yrow[r]) nq[r] += __expf(acc[r] * INV_T - m2[r]);
    }
    #pragma unroll
    for (int r = 0; r < 8; ++r) Dr[r] += redsum16(nq[r]);   // D = neg_raw + neg_q
  }

  // ---- pass 3: positive log-prob sum & count ----
  float psum[8] = {0.f, 0.f, 0.f, 0.f, 0.f, 0.f, 0.f, 0.f};
  float pcnt[8] = {0.f, 0.f, 0.f, 0.f, 0.f, 0.f, 0.f, 0.f};
  for (int t = 0; t < NA / 16; ++t) {
    int jcol = t * 16 + lm;
    int ycol = y[jcol];
    v8f acc = gemm16(a, Xn, t * 16, lm, half);
    #pragma unroll
    for (int r = 0; r < 8; ++r) {
      int i = rowBase + r + 8 * half;
      if (ycol == yrow[r] && jcol != i) {
        float l = acc[r] * INV_T - m1[r];
        psum[r] += l - __logf(__expf(l) + Dr[r]);
        pcnt[r] += 1.f;
      }
    }
  }
  #pragma unroll
  for (int r = 0; r < 8; ++r) { psum[r] = redsum16(psum[r]); pcnt[r] = redsum16(pcnt[r]); }

  // ---- cluster-center part: max pass then neg/pos pass (7 tiles) ----
  float m3[8];
  {
    float pm[8];
    #pragma unroll
    for (int r = 0; r < 8; ++r) pm[r] = -3.4e38f;
    for (int t = 0; t < NCCP / 16; ++t) {
      int j = t * 16 + lm;
      v8f acc = gemm16(a, Cn, t * 16, lm, half);
      if (j < NCC) {
        #pragma unroll
        for (int r = 0; r < 8; ++r) pm[r] = fmaxf(pm[r], acc[r] * INV_T);
      }
    }
    #pragma unroll
    for (int r = 0; r < 8; ++r) m3[r] = redmax16(pm[r]);
  }
  float negc[8] = {0.f, 0.f, 0.f, 0.f, 0.f, 0.f, 0.f, 0.f};
  float plog[8] = {0.f, 0.f, 0.f, 0.f, 0.f, 0.f, 0.f, 0.f};
  for (int t = 0; t < NCCP / 16; ++t) {
    int j = t * 16 + lm;
    bool jv = j < NCC;
    v8f acc = gemm16(a, Cn, t * 16, lm, half);
    #pragma unroll
    for (int r = 0; r < 8; ++r) {
      if (jv) {
        float l = acc[r] * INV_T - m3[r];
        if (j == yrow[r]) plog[r] += l;          // exactly one positive per row
        else             negc[r] += __expf(l);
      }
    }
  }
  #pragma unroll
  for (int r = 0; r < 8; ++r) { negc[r] = redsum16(negc[r]); plog[r] = redsum16(plog[r]); }

  // ---- write per-row outputs (one writer per row) ----
  if (lm == 0) {
    #pragma unroll
    for (int r = 0; r < 8; ++r) {
      int i = rowBase + r + 8 * half;
      float cnt = pcnt[r];
      float vld = cnt > 0.f ? 1.f : 0.f;
      float mlp = psum[r] / fmaxf(cnt, 1.f);
      per_ppc[i]  = -0.1f * mlp * vld;           // -(TEMP/BASE_TEMP) * mlpp * valid
      validArr[i] = vld;
      float pl = plog[r];
      mlppc[i] = pl - __logf(__expf(pl) + negc[r]);
    }
  }
}

// -------- deterministic final reduction --------
__global__ __launch_bounds__(256) void finalize_kernel(const float* __restrict__ per_ppc,
                                                       const float* __restrict__ validArr,
                                                       const float* __restrict__ mlppc,
                                                       float* __restrict__ out) {
  __shared__ float sa[256], sb[256], sc[256];
  int t = threadIdx.x;
  float a = 0.f, b = 0.f, c = 0.f;
  for (int i = t; i < NA; i += 256) { a += per_ppc[i]; b += validArr[i]; c += mlppc[i]; }
  sa[t] = a; sb[t] = b; sc[t] = c;
  __syncthreads();
  for (int s = 128; s > 0; s >>= 1) {
    if (t < s) { sa[t] += sa[t + s]; sb[t] += sb[t + s]; sc[t] += sc[t + s]; }
    __syncthreads();
  }
  if (t == 0) {
    float ppc = sa[0] / fmaxf(sb[0], 1.f);
    float pcc = -0.1f * sc[0] / (float)NA;
    out[0] = pcc * 10.f + ppc;   // loss
    out[1] = pcc;                // pcc_loss
    out[2] = ppc;                // ppc_loss
  }
}

extern "C" void kernel_launch(void* const* d_in, const int* in_sizes, int n_in,
                              void* d_out, int out_size, void* d_ws, size_t ws_size,
                              hipStream_t stream) {
  const float* X  = (const float*)d_in[0];   // X_anchor (4096,64)
  const float* PQ = (const float*)d_in[1];   // point_queue (102,150,64)
  const float* CC = (const float*)d_in[2];   // cluster_center (17,6,64)
  const int*   Y  = (const int*)d_in[3];     // y_anchor (4096,)

  float* ws   = (float*)d_ws;
  float* Xn   = ws;                          // 4096*64
  float* Qn   = Xn + (size_t)NA * DIMV;      // 15312*64 (padded)
  float* Cn   = Qn + (size_t)NQP * DIMV;     // 112*64   (padded)
  float* perp = Cn + (size_t)NCCP * DIMV;    // 4096
  float* vld  = perp + NA;                   // 4096
  float* mlp  = vld + NA;                    // 4096

  hipLaunchKernelGGL(norm_rows, dim3((NA  + 255) / 256), dim3(256), 0, stream, X,  Xn, NA,  NA);
  hipLaunchKernelGGL(norm_rows, dim3((NQP + 255) / 256), dim3(256), 0, stream, PQ, Qn, NQ,  NQP);
  hipLaunchKernelGGL(norm_rows, dim3(1),                 dim3(256), 0, stream, CC, Cn, NCC, NCCP);

  hipLaunchKernelGGL(stats_kernel, dim3(NA / 16), dim3(32), 0, stream,
                     Xn, Qn, Cn, Y, perp, vld, mlp);

  hipLaunchKernelGGL(finalize_kernel, dim3(1), dim3(256), 0, stream,
                     perp, vld, mlp, (float*)d_out);
}